// HistDRDoubleConv_40157944218403
// MI455X (gfx1250) — compile-verified
//
#include <hip/hip_runtime.h>
#include <hip/hip_bf16.h>
#include <stdint.h>

// HistDRDoubleConv for MI455X (gfx1250), fp32 end-to-end.
// Dense 64-ch dynamic conv as GEMM [64x72]@[72x65536] per sample via
// V_WMMA_F32_16X16X4_F32; A-panel async-copied to LDS; B built as an
// im2col panel in LDS so the WMMA loop is pure ds_load + wmma.

typedef __attribute__((ext_vector_type(2))) float v2f;
typedef __attribute__((ext_vector_type(8))) float v8f;

#define HW 65536      // 256*256
#define CH 8
#define RNUM 8

// ---------------------------------------------------------------- regionmap
__global__ __launch_bounds__(256)
void hdc_regionmap(const float* __restrict__ hist, unsigned char* __restrict__ map) {
    int i = blockIdx.x * 256 + threadIdx.x;          // i in [0, B*HW)
    int b = i >> 16;
    int pix = i & (HW - 1);
    const float* hb = hist + (size_t)b * RNUM * HW + pix;
    float best = hb[0];
    int r = 0;
    #pragma unroll
    for (int rr = 1; rr < RNUM; ++rr) {
        float v = hb[(size_t)rr * HW];
        if (v > best) { best = v; r = rr; }          // strict > keeps first max (jnp.argmax)
    }
    map[i] = (unsigned char)r;
}

// ---------------------------------------------------------------- avgpool 3x3
// window k=86, stride s=85 over 256x256 -> exact 3x3
__global__ __launch_bounds__(256)
void hdc_avgpool3(const float* __restrict__ src, float* __restrict__ p) {
    __shared__ float red[256];
    int win = blockIdx.x;          // 0..8
    int c   = blockIdx.y;          // 0..7
    int b   = blockIdx.z;          // 0..7
    int sy = (win / 3) * 85, sx = (win % 3) * 85;
    const float* sb = src + ((size_t)(b * CH + c)) * HW;
    float s = 0.0f;
    for (int e = threadIdx.x; e < 86 * 86; e += 256) {
        int ey = e / 86, ex = e - ey * 86;
        s += sb[(sy + ey) * 256 + sx + ex];
    }
    red[threadIdx.x] = s;
    __syncthreads();
    for (int st = 128; st > 0; st >>= 1) {
        if (threadIdx.x < (unsigned)st) red[threadIdx.x] += red[threadIdx.x + st];
        __syncthreads();
    }
    if (threadIdx.x == 0) p[(b * CH + c) * 9 + win] = red[0] * (1.0f / 7396.0f);
}

// ---------------------------------------------------------------- h = sigmoid(p@wa + ba)
__global__ __launch_bounds__(256)
void hdc_hmap(const float* __restrict__ p, const float* __restrict__ wa,
              const float* __restrict__ ba, float* __restrict__ h) {
    int i = blockIdx.x * 256 + threadIdx.x;          // [0, B*64*9)
    if (i >= 8 * 64 * 9) return;
    int win = i % 9;
    int o   = (i / 9) % 64;
    int b   = i / (9 * 64);
    float s = ba[o];
    #pragma unroll
    for (int c = 0; c < CH; ++c) s += p[(b * CH + c) * 9 + win] * wa[o * CH + c];
    h[i] = 1.0f / (1.0f + __expf(-s));
}

// ---------------------------------------------------------------- dynamic kernels
// A-matrix layout: ki[b][ro][k], ro in [0,64), k = (ky*3+kx)*8 + c  (64x72)
__global__ __launch_bounds__(256)
void hdc_genkern(const float* __restrict__ h, const float* __restrict__ wb,
                 const float* __restrict__ bb, float* __restrict__ ki) {
    int i = blockIdx.x * 256 + threadIdx.x;          // [0, B*64*72)
    if (i >= 8 * 64 * 72) return;
    int k  = i % 72;
    int ro = (i / 72) % 64;
    int b  = i / (72 * 64);
    int c  = k & 7;                                   // input channel
    int t9 = k >> 3;                                  // ky*3+kx
    int r   = ro >> 3;
    int o64 = ((ro & 7) << 3) | c;                    // reshape [R,Cin*out]->[R*out,Cin]
    float s = bb[r * 64 + o64];
    #pragma unroll
    for (int j = 0; j < RNUM; ++j)
        s += h[b * 576 + (r * 8 + j) * 9 + t9] * wb[(r * 64 + o64) * 8 + j];
    ki[i] = s;
}

// ---------------------------------------------------------------- WMMA conv + select + BN/ReLU
// grid: (512, B), block 256 (8 waves). Block covers 128 contiguous pixels of
// one image row (row = blockIdx.x>>1, colbase = (blockIdx.x&1)*128).
// A (64x72) async-copied to LDS; B staged as im2col panel Bs[72][128] in LDS.
// Inner loop: pure ds_load + 4x v_wmma_f32_16x16x4_f32 per K-step.
__global__ __launch_bounds__(256)
void hdc_drconv16(const float* __restrict__ src, const float* __restrict__ ki,
                  const unsigned char* __restrict__ map,
                  const float* __restrict__ g, const float* __restrict__ be,
                  const float* __restrict__ mn, const float* __restrict__ vr,
                  float* __restrict__ dst) {
    __shared__ alignas(16) float As[64 * 72];         // 18 KB  A[ro][k]
    __shared__ alignas(16) float Bs[72 * 128];        // 36 KB  B[k][px]
    const int b   = blockIdx.y;
    const int tid = threadIdx.x;
    const int row     = blockIdx.x >> 1;              // image row
    const int colbase = (blockIdx.x & 1) << 7;        // 0 or 128

    // ---- async-copy A panel (4608 floats = 1152 x b128) into LDS
    const float* kib = ki + (size_t)b * 64 * 72;
    for (int i = tid; i < 1152; i += 256) {
        uint32_t loff  = (uint32_t)(uintptr_t)(const void*)As + (uint32_t)i * 16u;
        uint64_t gaddr = (uint64_t)(uintptr_t)kib + (uint64_t)i * 16u;
        asm volatile("global_load_async_to_lds_b128 %0, %1, off"
                     :: "v"(loff), "v"(gaddr) : "memory");
    }

    // ---- stage im2col panel: Bs[k][px] = x[b, c, row+ky-1, colbase+px+kx-1]
    const float* sb = src + (size_t)b * CH * HW;
    for (int idx = tid; idx < 72 * 128; idx += 256) { // 36 iters
        int k  = idx >> 7;
        int px = idx & 127;
        int c  = k & 7;
        int t9 = k >> 3;                              // 0..8
        int ky = (t9 * 11) >> 5;                      // t9/3 without divide
        int kx = t9 - ky * 3;
        int yy = row + ky - 1;
        int xs = colbase + px + kx - 1;
        float val = 0.0f;
        if ((unsigned)yy < 256u && (unsigned)xs < 256u)
            val = sb[c * HW + yy * 256 + xs];
        Bs[k * 128 + px] = val;
    }

    asm volatile("s_wait_asynccnt 0x0" ::: "memory"); // A copy visible in LDS
    __syncthreads();

    const int wave = tid >> 5;
    const int lane = tid & 31;
    const int n    = lane & 15;                       // pixel within wave's subtile
    const bool hi  = lane >= 16;                      // high half-wave: K+2,K+3
    const int cl   = (wave << 4) | n;                 // pixel column within block panel

    v8f acc0 = {}; v8f acc1 = {}; v8f acc2 = {}; v8f acc3 = {};

    for (int kk = 0; kk < 18; ++kk) {
        const int K0 = kk * 4 + (hi ? 2 : 0);
        v2f bf;
        bf.x = Bs[K0 * 128 + cl];
        bf.y = Bs[(K0 + 1) * 128 + cl];

        const int ab = n * 72 + K0;                   // A row M = m*16+n, cols K0,K0+1
        v2f a0, a1, a2, a3;
        a0.x = As[ab];             a0.y = As[ab + 1];
        a1.x = As[ab + 16 * 72];   a1.y = As[ab + 16 * 72 + 1];
        a2.x = As[ab + 32 * 72];   a2.y = As[ab + 32 * 72 + 1];
        a3.x = As[ab + 48 * 72];   a3.y = As[ab + 48 * 72 + 1];

        acc0 = __builtin_amdgcn_wmma_f32_16x16x4_f32(false, a0, false, bf, (short)0, acc0, false, false);
        acc1 = __builtin_amdgcn_wmma_f32_16x16x4_f32(false, a1, false, bf, (short)0, acc1, false, false);
        acc2 = __builtin_amdgcn_wmma_f32_16x16x4_f32(false, a2, false, bf, (short)0, acc2, false, false);
        acc3 = __builtin_amdgcn_wmma_f32_16x16x4_f32(false, a3, false, bf, (short)0, acc3, false, false);
    }

    // Epilogue: per-pixel region select + BN + ReLU.
    // Channels r*8+j live in M-tile msel=r>>1, half (r&1): low lanes hold rows
    // 0..7 (VGPR j), high lanes rows 8..15 (VGPR j). Channel-in-slice == j.
    const int xx   = colbase + cl;
    const int pix  = row * 256 + xx;
    const int r    = map[(size_t)b * HW + pix];
    const bool act = ((r & 1) != 0) == hi;
    const int msel = r >> 1;
    float* db = dst + (size_t)b * CH * HW + pix;

    #pragma unroll
    for (int m = 0; m < 4; ++m) {
        if (act && msel == m) {
            v8f a = (m == 0) ? acc0 : (m == 1) ? acc1 : (m == 2) ? acc2 : acc3;
            #pragma unroll
            for (int j = 0; j < 8; ++j) {
                float sc = g[j] * __frsqrt_rn(vr[j] + 1e-5f);
                float o  = (a[j] - mn[j]) * sc + be[j];
                db[j * HW] = o > 0.0f ? o : 0.0f;
            }
        }
    }
}

// ---------------------------------------------------------------- launcher
extern "C" void kernel_launch(void* const* d_in, const int* in_sizes, int n_in,
                              void* d_out, int out_size, void* d_ws, size_t ws_size,
                              hipStream_t stream) {
    const float* x    = (const float*)d_in[0];
    const float* hist = (const float*)d_in[1];
    const float* w1a  = (const float*)d_in[2];
    const float* b1a  = (const float*)d_in[3];
    const float* w1b  = (const float*)d_in[4];
    const float* b1b  = (const float*)d_in[5];
    const float* g1   = (const float*)d_in[6];
    const float* be1  = (const float*)d_in[7];
    const float* m1   = (const float*)d_in[8];
    const float* v1   = (const float*)d_in[9];
    const float* w2a  = (const float*)d_in[10];
    const float* b2a  = (const float*)d_in[11];
    const float* w2b  = (const float*)d_in[12];
    const float* b2b  = (const float*)d_in[13];
    const float* g2   = (const float*)d_in[14];
    const float* be2  = (const float*)d_in[15];
    const float* m2   = (const float*)d_in[16];
    const float* v2   = (const float*)d_in[17];

    char* ws = (char*)d_ws;
    unsigned char* map = (unsigned char*)ws;                 // 512 KB
    float* p    = (float*)(ws + 0x80000);                    // 2.25 KB
    float* hbuf = (float*)(ws + 0x81000);                    // 18 KB
    float* ki   = (float*)(ws + 0x86000);                    // 144 KB
    float* y1   = (float*)(ws + 0xAB000);                    // 16 MB
    float* out  = (float*)d_out;

    hdc_regionmap<<<2048, 256, 0, stream>>>(hist, map);

    // Layer 1
    hdc_avgpool3<<<dim3(9, 8, 8), 256, 0, stream>>>(x, p);
    hdc_hmap<<<18, 256, 0, stream>>>(p, w1a, b1a, hbuf);
    hdc_genkern<<<144, 256, 0, stream>>>(hbuf, w1b, b1b, ki);
    hdc_drconv16<<<dim3(512, 8), 256, 0, stream>>>(x, ki, map, g1, be1, m1, v1, y1);

    // Layer 2 (kernels depend on y1)
    hdc_avgpool3<<<dim3(9, 8, 8), 256, 0, stream>>>(y1, p);
    hdc_hmap<<<18, 256, 0, stream>>>(p, w2a, b2a, hbuf);
    hdc_genkern<<<144, 256, 0, stream>>>(hbuf, w2b, b2b, ki);
    hdc_drconv16<<<dim3(512, 8), 256, 0, stream>>>(y1, ki, map, g2, be2, m2, v2, out);
}